// TrueShadowlessEmbedding_40518721471147
// MI455X (gfx1250) — compile-verified
//
#include <hip/hip_runtime.h>

// Embedding LUT-dequant: out[t, d] = lut[ base_idx[ids[t], d] * 256 + fine_idx[ids[t], d] ]
//
// MI455X strategy:
//  * 256 KB LUT lives entirely in LDS (CDNA5: 320 KB/WGP) -> 64M random 4B
//    lookups become ds_load_b32 bank-gathers instead of uncoalesced vmem.
//  * LUT staged via gfx1250 async global->LDS copies (ASYNCcnt) overlapped
//    with token-id staging; drained with s_wait_asynccnt.
//  * base/fine index rows and the output are moved with coalesced b128
//    vector ops; rows two iterations ahead are prefetched (global_prefetch).
//  * HBM floor: ~128 MiB index reads + 64 MiB writes ~= 8.3 us @ 23.3 TB/s.

#define DIM          1024
#define LUT_SIZE     65536
#define BLOCK_T      1024          // 32 wave32s -> fills the WGP at 1 WG/WGP
#define TOK_PER_BLK  128

// Types matching the async-to-LDS builtin signature (leaked by the round-1
// diagnostic): param0 is `int __attribute__((vector_size(16))) __device__ *`.
typedef int v4i __attribute__((vector_size(16)));
typedef __attribute__((address_space(1))) v4i* g4p;   // global (AS1)
typedef __attribute__((address_space(3))) v4i* l4p;   // LDS (AS3)

__global__ __launch_bounds__(BLOCK_T, 1)
void embed_dequant_kernel(const int*   __restrict__ ids,
                          const int*   __restrict__ base_idx,
                          const int*   __restrict__ fine_idx,
                          const float* __restrict__ lut,
                          float*       __restrict__ out,
                          int n_tokens)
{
    __shared__ __align__(16) float lds_lut[LUT_SIZE];   // 256 KB (<= 320 KB/WGP)
    __shared__ int lds_ids[TOK_PER_BLK];

    const int tid  = threadIdx.x;
    const int tok0 = blockIdx.x * TOK_PER_BLK;

    // ---- Stage the full 256 KB LUT into LDS ----
#if defined(__AMDGCN__) && \
    __has_builtin(__builtin_amdgcn_global_load_async_to_lds_b128) && \
    __has_builtin(__builtin_amdgcn_s_wait_asynccnt)
    // gfx1250 async DMA path: 16 B per lane per issue, tracked by ASYNCcnt.
    for (int i = tid; i < LUT_SIZE / 4; i += BLOCK_T) {
        __builtin_amdgcn_global_load_async_to_lds_b128(
            (g4p)(lut + 4 * i),
            (l4p)(&lds_lut[4 * i]),
            /*imm offset*/ 0, /*cpol*/ 0);
    }
    // Overlap: stage this block's token ids while async copies are in flight.
    for (int i = tid; i < TOK_PER_BLK; i += BLOCK_T) {
        const int t = tok0 + i;
        lds_ids[i] = (t < n_tokens) ? ids[t] : 0;
    }
    __builtin_amdgcn_s_wait_asynccnt(0);
#else
    for (int i = tid; i < LUT_SIZE; i += BLOCK_T) lds_lut[i] = lut[i];
    for (int i = tid; i < TOK_PER_BLK; i += BLOCK_T) {
        const int t = tok0 + i;
        lds_ids[i] = (t < n_tokens) ? ids[t] : 0;
    }
#endif
    __syncthreads();

    // ---- Main gather loop: 4 tokens per iteration ----
    // 256 threads per token, 4 consecutive dims per thread -> b128 loads/stores.
    const int sub  = tid >> 8;           // 0..3 : token within the group of 4
    const int dofs = (tid & 255) << 2;   // 0..1020 : dim offset

    for (int i = 0; i < TOK_PER_BLK; i += 4) {
        const int t = tok0 + i + sub;
        if (t < n_tokens) {
            const int    row   = lds_ids[i + sub];
            const size_t rbase = (size_t)row * DIM + dofs;

            // Prefetch index rows two iterations ahead (gfx1250 global_prefetch).
            if (i + 8 < TOK_PER_BLK) {
                const int    prow = lds_ids[i + 8 + sub];
                const size_t pb   = (size_t)prow * DIM + dofs;
                __builtin_prefetch(base_idx + pb, 0, 1);
                __builtin_prefetch(fine_idx + pb, 0, 1);
            }

            const int4 b = *(const int4*)(base_idx + rbase);
            const int4 f = *(const int4*)(fine_idx + rbase);

            float4 r;
            r.x = lds_lut[(b.x << 8) + f.x];
            r.y = lds_lut[(b.y << 8) + f.y];
            r.z = lds_lut[(b.z << 8) + f.z];
            r.w = lds_lut[(b.w << 8) + f.w];

            *(float4*)(out + (size_t)t * DIM + dofs) = r;
        }
    }
}

extern "C" void kernel_launch(void* const* d_in, const int* in_sizes, int n_in,
                              void* d_out, int out_size, void* d_ws, size_t ws_size,
                              hipStream_t stream)
{
    (void)n_in; (void)out_size; (void)d_ws; (void)ws_size;

    const int*   ids  = (const int*)  d_in[0];   // (B, S) int32
    const int*   base = (const int*)  d_in[1];   // (V, D) int32
    const int*   fine = (const int*)  d_in[2];   // (V, D) int32
    const float* lut  = (const float*)d_in[3];   // (65536,) f32
    float*       out  = (float*)      d_out;     // (B, S, D) f32

    const int n_tokens = in_sizes[0];                                  // 16384
    const int nblk     = (n_tokens + TOK_PER_BLK - 1) / TOK_PER_BLK;   // 128

    embed_dequant_kernel<<<nblk, BLOCK_T, 0, stream>>>(ids, base, fine, lut, out, n_tokens);
}